// IO_16569983828129
// MI455X (gfx1250) — compile-verified
//
#include <hip/hip_runtime.h>
#include <math.h>

// ---------------------------------------------------------------------------
// Label-smoothed cross-entropy over vocab, streaming (no logits tensor).
//   B*S = 4096 tokens, D = 1024, V = 50257, LABEL_SMOOTH = 0.1
// Kernel 1: fp32 WMMA GEMM tiles (v_wmma_f32_16x16x4_f32), W tile resident in
//   LDS (once per slab from HBM), X double-buffered via async LDS copies
//   (global_load_async_to_lds_b128 / s_wait_asynccnt), per-token
//   online-softmax partials reduced in-wave with xor-shuffles.
// Kernel 2: per-token log-sum-exp merge + masked reduction to scalar.
// ---------------------------------------------------------------------------

typedef __attribute__((ext_vector_type(2))) float v2f;
typedef __attribute__((ext_vector_type(8))) float v8f;
typedef int v4i_vs __attribute__((vector_size(4 * sizeof(int))));   // b128 unit

#define DMODEL      1024
#define VOCAB       50257
#define NTOK        4096            // B*S
#define COLS_PER_WG 128
#define SLAB        32              // cols resident in LDS at a time
#define NSLAB       (COLS_PER_WG / SLAB)
#define NWG         ((VOCAB + COLS_PER_WG - 1) / COLS_PER_WG)   // 393
#define NWGV        (NWG * 2)       // one partial per wave (64 cols each: 4 slabs x 16)
#define PITCH       1028            // LDS row pitch (floats): 16B-aligned, bank-spread
#define NRB         (NTOK / 16)     // 256 row blocks of 16 tokens

// LDS floats: sW[32][PITCH] + sX[2][16][PITCH] + sT[2][16] (ints) ~ 257 KB
#define LDS_W       (SLAB * PITCH)
#define LDS_X       (16 * PITCH)
#define LDS_FLOATS  (LDS_W + 2 * LDS_X + 2 * 16)

// ---- CDNA5 async global->LDS copy (ASYNCcnt path), guarded fallback ----
__device__ __forceinline__ void async_cp_b128(void* lds_dst, const void* gsrc) {
#if defined(__has_builtin) && __has_builtin(__builtin_amdgcn_global_load_async_to_lds_b128)
    __builtin_amdgcn_global_load_async_to_lds_b128(
        (__attribute__((address_space(1))) v4i_vs*)(gsrc),
        (__attribute__((address_space(3))) v4i_vs*)(lds_dst),
        /*offset=*/0, /*cpol=*/0);
#else
    *(float4*)lds_dst = *(const float4*)gsrc;   // synchronous fallback
#endif
}

__device__ __forceinline__ void wait_async_all() {
#if defined(__has_builtin) && __has_builtin(__builtin_amdgcn_s_wait_asynccnt)
    __builtin_amdgcn_s_wait_asynccnt(0);
#else
    asm volatile("s_wait_asynccnt 0x0" ::: "memory");
#endif
}

extern "C" __global__ void __launch_bounds__(64, 1)
ce_partial(const int* __restrict__ targets,
           const float* __restrict__ X,      // [NTOK][DMODEL]
           const float* __restrict__ W,      // [VOCAB][DMODEL]
           float* __restrict__ P,            // [NWGV][NTOK][3] : m, l, sum_z
           float* __restrict__ ZT)           // [NTOK] : z[target]
{
    extern __shared__ float lds[];
    float* sW  = lds;                 // [SLAB][PITCH]
    float* sX0 = lds + LDS_W;         // [16][PITCH]  (double buffer 0)
    float* sX1 = sX0 + LDS_X;         // [16][PITCH]  (double buffer 1)
    int*   sT  = (int*)(sX1 + LDS_X); // [2][16] staged targets

    const int tid  = threadIdx.x;
    const int wave = tid >> 5;                // 0..1 (wave32)
    const int lane = tid & 31;
    const int half = lane >> 4;               // 0: lanes 0-15, 1: lanes 16-31
    const int l16  = lane & 15;
    const int wg   = blockIdx.x;

    for (int slab = 0; slab < NSLAB; ++slab) {
        const int  colBase = wg * COLS_PER_WG + slab * SLAB;  // WG's 32 cols
        const int  myCol   = colBase + wave * 16 + l16;       // this lane's col
        const bool colOK   = myCol < VOCAB;

        __syncthreads();   // previous slab's WMMA readers done before sW rewrite

        // ---- Stage W tile: 32 rows x 1024 f32, coalesced, zero-filled tail ----
        for (int i = tid; i < SLAB * (DMODEL / 4); i += 64) {
            const int r = i >> 8;             // / 256 chunks per row
            const int c = (i & 255) << 2;
            const int v = colBase + r;
            float4 w4 = make_float4(0.f, 0.f, 0.f, 0.f);
            if (v < VOCAB) w4 = *(const float4*)&W[(size_t)v * DMODEL + c];
            *(float4*)&sW[r * PITCH + c] = w4;
        }

        // ---- Prologue: async-stage X block 0 into buffer 0, targets too ----
        for (int it = 0; it < 64; ++it) {
            const int c  = it * 64 + tid;     // 16-byte chunk id in 16x1024 block
            const int r  = c >> 8;
            const int c4 = (c & 255) << 2;
            async_cp_b128(&sX0[r * PITCH + c4], &X[c << 2]);
        }
        if (tid < 16) sT[tid] = targets[tid];

        for (int rb = 0; rb < NRB; ++rb) {
            float*     sXc = (rb & 1) ? sX1 : sX0;
            float*     sXn = (rb & 1) ? sX0 : sX1;
            const int* sTc = sT + (rb & 1) * 16;
            int*       sTn = sT + ((rb + 1) & 1) * 16;

            wait_async_all();       // this wave's copies into sXc complete
            __syncthreads();        // all waves' copies visible

            // ---- Hoist this block's 8 staged targets into registers early; --
            // the dscnt wait is absorbed under the WMMA loop below.
            const int4 t0 = *(const int4*)&sTc[8 * half];
            const int4 t1 = *(const int4*)&sTc[8 * half + 4];
            const int  tg[8] = { t0.x, t0.y, t0.z, t0.w, t1.x, t1.y, t1.z, t1.w };

            // ---- Issue next block's async copies (overlaps WMMA below) ----
            if (rb + 1 < NRB) {
                const float* Xblk = X + (size_t)(rb + 1) * 16 * DMODEL;
                for (int it = 0; it < 64; ++it) {
                    const int c  = it * 64 + tid;
                    const int r  = c >> 8;
                    const int c4 = (c & 255) << 2;
                    async_cp_b128(&sXn[r * PITCH + c4], &Xblk[c << 2]);
                }
                if (tid < 16) sTn[tid] = targets[(rb + 1) * 16 + tid];
            }

            // ---- 16x16 fp32 WMMA tile over K=1024 ----
            // A frag (16x4): lane l16 = row M; K pair {0,1} / {2,3} by half.
            // B frag (4x16): lane l16 = col N; same K split by half.
            const float* aRow = &sXc[l16 * PITCH + 2 * half];
            const float* bRow = &sW[(wave * 16 + l16) * PITCH + 2 * half];
            v8f acc = {};
#pragma unroll 8
            for (int k = 0; k < DMODEL; k += 4) {
                const float2 av = *(const float2*)&aRow[k];
                const float2 bv = *(const float2*)&bRow[k];
                v2f a; a.x = av.x; a.y = av.y;
                v2f b; b.x = bv.x; b.y = bv.y;
                acc = __builtin_amdgcn_wmma_f32_16x16x4_f32(
                          false, a, false, b, (short)0, acc, false, false);
            }

            // ---- In-wave per-row stats (C: VGPR j = row j+8*half, lane&15 = col)
            // xor-shuffles with masks 1,2,4,8 reduce within each 16-lane half.
            float rm[8], rl[8], rs[8];
#pragma unroll
            for (int j = 0; j < 8; ++j) {
                float z = colOK ? acc[j] : -INFINITY;
                z = fmaxf(z, __shfl_xor(z, 1, 32));
                z = fmaxf(z, __shfl_xor(z, 2, 32));
                z = fmaxf(z, __shfl_xor(z, 4, 32));
                z = fmaxf(z, __shfl_xor(z, 8, 32));
                float e = colOK ? __expf(acc[j] - z) : 0.f;
                e += __shfl_xor(e, 1, 32);
                e += __shfl_xor(e, 2, 32);
                e += __shfl_xor(e, 4, 32);
                e += __shfl_xor(e, 8, 32);
                float s = colOK ? acc[j] : 0.f;
                s += __shfl_xor(s, 1, 32);
                s += __shfl_xor(s, 2, 32);
                s += __shfl_xor(s, 4, 32);
                s += __shfl_xor(s, 8, 32);
                rm[j] = z; rl[j] = e; rs[j] = s;
            }

            // ---- z[target]: unique owner lane stores it (registers only) ----
#pragma unroll
            for (int j = 0; j < 8; ++j) {
                if (tg[j] == myCol) ZT[rb * 16 + j + 8 * half] = acc[j];
            }

            // ---- Merge this wave's 16-col partial into P (lane l16<8 = row) --
            float mm = rm[0], ll = rl[0], ss = rs[0];
#pragma unroll
            for (int j = 1; j < 8; ++j) {
                const bool p = (l16 == j);
                mm = p ? rm[j] : mm;
                ll = p ? rl[j] : ll;
                ss = p ? rs[j] : ss;
            }
            if (l16 < 8) {
                const int    t  = rb * 16 + l16 + 8 * half;
                const size_t pi = (((size_t)(wg * 2 + wave)) * NTOK + t) * 3;
                if (slab == 0) {              // first touch each call
                    P[pi] = mm; P[pi + 1] = ll; P[pi + 2] = ss;
                } else {                      // online log-sum-exp merge
                    const float m0 = P[pi], l0 = P[pi + 1], s0 = P[pi + 2];
                    const float mn = fmaxf(m0, mm);
                    P[pi]     = mn;
                    P[pi + 1] = l0 * __expf(m0 - mn) + ll * __expf(mm - mn);
                    P[pi + 2] = s0 + ss;
                }
            }
        }
    }
}

extern "C" __global__ void __launch_bounds__(256, 1)
ce_finalize(const unsigned char* __restrict__ masks,   // jax bool = 1 byte
            const float* __restrict__ P,
            const float* __restrict__ ZT,
            float* __restrict__ out)
{
    __shared__ float rA[256], rB[256], rN[256];
    const int tid = threadIdx.x;
    float accA = 0.f, accB = 0.f, accN = 0.f;

    for (int t = tid; t < NTOK; t += 256) {
        float m = -INFINITY, l = 0.f, s = 0.f;
        for (int w = 0; w < NWGV; ++w) {
            const size_t pi = ((size_t)w * NTOK + t) * 3;
            const float mi = P[pi], li = P[pi + 1], si = P[pi + 2];
            const float mn = fmaxf(m, mi);
            l = l * __expf(m - mn) + li * __expf(mi - mn);
            m = mn;
            s += si;
        }
        const float logZ     = m + __logf(l);
        const float tok_logp = ZT[t] - logZ;                   // logp[target]
        const float sum_logp = s - (float)VOCAB * logZ;        // sum_v logp[v]
        const float mk = masks[t] ? 1.f : 0.f;
        accA += mk * tok_logp;
        accB += mk * sum_logp;
        accN += mk;
    }
    rA[tid] = accA; rB[tid] = accB; rN[tid] = accN;
    __syncthreads();
    for (int st = 128; st > 0; st >>= 1) {
        if (tid < st) {
            rA[tid] += rA[tid + st];
            rB[tid] += rB[tid + st];
            rN[tid] += rN[tid + st];
        }
        __syncthreads();
    }
    if (tid == 0) {
        const float nll       = -rA[0] / rN[0];
        const float mean_logp =  rB[0] / (rN[0] * (float)VOCAB);
        out[0] = nll * 0.9f - mean_logp * 0.1f;   // LABEL_SMOOTH = 0.1
    }
}

extern "C" void kernel_launch(void* const* d_in, const int* in_sizes, int n_in,
                              void* d_out, int out_size, void* d_ws, size_t ws_size,
                              hipStream_t stream)
{
    const int*           targets = (const int*)d_in[0];
    const unsigned char* masks   = (const unsigned char*)d_in[1];
    const float*         X       = (const float*)d_in[2];   // outputs [2,2048,1024]
    const float*         W       = (const float*)d_in[3];   // [50257,1024]

    float* P  = (float*)d_ws;                        // NWGV*NTOK*3 floats (~38.6 MB)
    float* ZT = P + (size_t)NWGV * NTOK * 3;         // NTOK floats

    const size_t ldsBytes = (size_t)LDS_FLOATS * sizeof(float);  // ~257 KB

    ce_partial<<<NWG, 64, ldsBytes, stream>>>(targets, X, W, P, ZT);
    ce_finalize<<<1, 256, 0, stream>>>(masks, P, ZT, (float*)d_out);
}